// HingeLoss_2224793059461
// MI455X (gfx1250) — compile-verified
//
#include <hip/hip_runtime.h>
#include <math.h>

typedef float v2f __attribute__((ext_vector_type(2)));
typedef float v8f __attribute__((ext_vector_type(8)));

#define TILE_M 16
#define WAVES  4
#define BLOCK  (WAVES * 32)

// ---------------------------------------------------------------------------
// Kernel 1: per-16-row tile, WMMA-based K reduction.
//   c1 = out x tgt^T   (diag -> dot_pos)
//   c2 = out x tgtj^T  (diag -> dot_neg)
//   c3 = out x out^T   (diag -> ||out||^2)
//   c4 = tgt x tgt^T   (diag -> ||tgt||^2)
// ---------------------------------------------------------------------------
__global__ __launch_bounds__(BLOCK) void hinge_rowstats_wmma(
    const float* __restrict__ outp, const float* __restrict__ tgtp,
    const int* __restrict__ jidx, int D,
    float* __restrict__ dpos, float* __restrict__ dneg,
    float* __restrict__ sqo,  float* __restrict__ sqt,
    int* __restrict__ samef)
{
    __shared__ float tiles[WAVES][4][TILE_M * TILE_M];
    __shared__ int   sameAll[TILE_M];

    const int tid  = threadIdx.x;
    const int wave = tid >> 5;
    const int lane = tid & 31;
    const int r    = lane & 15;   // row within 16-row tile
    const int h    = lane >> 4;   // lane half (K sub-pair selector)
    const int rowG = blockIdx.x * TILE_M + r;

    if (tid < TILE_M) sameAll[tid] = ~0;
    __syncthreads();

    const float* po = outp + (size_t)rowG * D;
    const float* pt = tgtp + (size_t)rowG * D;
    const float* pg = tgtp + (size_t)jidx[rowG] * D;

    const int Dq   = D >> 2;                 // K range per wave
    const int kb   = wave * Dq + 2 * h;      // per-lane K base
    const int iter = Dq >> 3;                // K=8 per iteration

    v8f c1 = {}, c2 = {}, c3 = {}, c4 = {};
    int ok = ~0;

    #pragma unroll 2
    for (int it = 0; it < iter; ++it) {
        const int k = kb + it * 8;
        // Each float2 is simultaneously an A-operand (rows) and a B-operand
        // (transposed columns) for the 16x16x4 f32 WMMA layout.
        v2f o_lo = *(const v2f*)(po + k);
        v2f o_hi = *(const v2f*)(po + k + 4);
        v2f t_lo = *(const v2f*)(pt + k);
        v2f t_hi = *(const v2f*)(pt + k + 4);
        v2f g_lo = *(const v2f*)(pg + k);
        v2f g_hi = *(const v2f*)(pg + k + 4);

        // torch.allclose(t_j, t_i): |a-b| <= atol + rtol*|b|, b = t_i
        ok &= (fabsf(g_lo.x - t_lo.x) <= 1e-8f + 1e-5f * fabsf(t_lo.x)) ? ~0 : 0;
        ok &= (fabsf(g_lo.y - t_lo.y) <= 1e-8f + 1e-5f * fabsf(t_lo.y)) ? ~0 : 0;
        ok &= (fabsf(g_hi.x - t_hi.x) <= 1e-8f + 1e-5f * fabsf(t_hi.x)) ? ~0 : 0;
        ok &= (fabsf(g_hi.y - t_hi.y) <= 1e-8f + 1e-5f * fabsf(t_hi.y)) ? ~0 : 0;

        c1 = __builtin_amdgcn_wmma_f32_16x16x4_f32(false, o_lo, false, t_lo, (short)0, c1, false, false);
        c2 = __builtin_amdgcn_wmma_f32_16x16x4_f32(false, o_lo, false, g_lo, (short)0, c2, false, false);
        c3 = __builtin_amdgcn_wmma_f32_16x16x4_f32(false, o_lo, false, o_lo, (short)0, c3, false, false);
        c4 = __builtin_amdgcn_wmma_f32_16x16x4_f32(false, t_lo, false, t_lo, (short)0, c4, false, false);
        c1 = __builtin_amdgcn_wmma_f32_16x16x4_f32(false, o_hi, false, t_hi, (short)0, c1, false, false);
        c2 = __builtin_amdgcn_wmma_f32_16x16x4_f32(false, o_hi, false, g_hi, (short)0, c2, false, false);
        c3 = __builtin_amdgcn_wmma_f32_16x16x4_f32(false, o_hi, false, o_hi, (short)0, c3, false, false);
        c4 = __builtin_amdgcn_wmma_f32_16x16x4_f32(false, t_hi, false, t_hi, (short)0, c4, false, false);
    }

    // Spill C tiles to LDS: element (M,N) -> M*16+N ; M = vgpr + (lane<16?0:8)
    const int mofs = (lane < 16) ? 0 : 8;
    const int n    = lane & 15;
    #pragma unroll
    for (int v = 0; v < 8; ++v) {
        const int idx = (mofs + v) * TILE_M + n;
        tiles[wave][0][idx] = c1[v];
        tiles[wave][1][idx] = c2[v];
        tiles[wave][2][idx] = c3[v];
        tiles[wave][3][idx] = c4[v];
    }
    atomicAnd(&sameAll[r], ok);
    __syncthreads();

    // Threads 0..15: sum diagonals across the 4 waves, write row stats.
    if (tid < TILE_M) {
        const int dg = tid * (TILE_M + 1);
        float s1 = 0.f, s2 = 0.f, s3 = 0.f, s4 = 0.f;
        #pragma unroll
        for (int w = 0; w < WAVES; ++w) {
            s1 += tiles[w][0][dg];
            s2 += tiles[w][1][dg];
            s3 += tiles[w][2][dg];
            s4 += tiles[w][3][dg];
        }
        const int g = blockIdx.x * TILE_M + tid;
        dpos[g]  = s1;
        dneg[g]  = s2;
        sqo[g]   = s3;
        sqt[g]   = s4;
        samef[g] = (sameAll[tid] != 0);
    }
}

// ---------------------------------------------------------------------------
// Kernel 2: gather partner norm, cosine hinge, mean. Single block -> no
// global atomics and no dependence on d_out initialization.
// ---------------------------------------------------------------------------
__global__ __launch_bounds__(256) void hinge_finalize(
    const float* __restrict__ dpos, const float* __restrict__ dneg,
    const float* __restrict__ sqo,  const float* __restrict__ sqt,
    const int* __restrict__ samef,  const int* __restrict__ jidx,
    int B, float* __restrict__ out)
{
    __shared__ float red[256];
    float s = 0.f;
    for (int i = threadIdx.x; i < B; i += 256) {
        float no  = fmaxf(sqrtf(sqo[i]), 1e-6f);
        float nt  = fmaxf(sqrtf(sqt[i]), 1e-6f);
        float ntj = fmaxf(sqrtf(sqt[jidx[i]]), 1e-6f);
        float cp  = dpos[i] / (nt * no);
        float cn  = dneg[i] / (ntj * no);
        float m   = samef[i] ? -0.5f : 0.5f;
        s += fmaxf(m - cp + cn, 0.f);
    }
    red[threadIdx.x] = s;
    __syncthreads();
    for (int st = 128; st > 0; st >>= 1) {
        if (threadIdx.x < st) red[threadIdx.x] += red[threadIdx.x + st];
        __syncthreads();
    }
    if (threadIdx.x == 0) out[0] = red[0] / (float)B;
}

// ---------------------------------------------------------------------------
extern "C" void kernel_launch(void* const* d_in, const int* in_sizes, int n_in,
                              void* d_out, int out_size, void* d_ws, size_t ws_size,
                              hipStream_t stream) {
    const float* outp = (const float*)d_in[0];
    const float* tgtp = (const float*)d_in[1];
    const int*   jidx = (const int*)d_in[2];
    const int B = in_sizes[2];
    const int D = in_sizes[0] / B;

    float* ws   = (float*)d_ws;
    float* dpos = ws;
    float* dneg = ws + (size_t)B;
    float* sqo  = ws + (size_t)2 * B;
    float* sqt  = ws + (size_t)3 * B;
    int*   same = (int*)(ws + (size_t)4 * B);

    hinge_rowstats_wmma<<<B / TILE_M, BLOCK, 0, stream>>>(
        outp, tgtp, jidx, D, dpos, dneg, sqo, sqt, same);
    hinge_finalize<<<1, 256, 0, stream>>>(
        dpos, dneg, sqo, sqt, same, jidx, B, (float*)d_out);
}